// multihead_attention_40492951667373
// MI455X (gfx1250) — compile-verified
//
#include <hip/hip_runtime.h>
#include <math.h>

// MI455X / gfx1250, wave32. fp32 WMMA 16x16x4 matches head_dim=4 exactly.

typedef __attribute__((ext_vector_type(2))) float v2f;
typedef __attribute__((ext_vector_type(8))) float v8f;

#define BB 4
#define SS 2048
#define DD 64
#define HH 16
#define HDIM 4

static __device__ __forceinline__ v8f wmma_f32_16x16x4(v2f a, v2f b, v8f c) {
  // 8 args: (neg_a, A, neg_b, B, c_mod, C, reuse_a, reuse_b)
  return __builtin_amdgcn_wmma_f32_16x16x4_f32(false, a, false, b, (short)0, c,
                                               false, false);
}

// ---------------------------------------------------------------------------
// Kernel 1: Q/K/V = relu(x @ W + b).  One wave = one 16-row x 64-col tile.
// 4 waves per block. Tiles: 3 matrices * (B*S/16) = 1536 -> 384 blocks.
// Fully unrolled K-loop: afrag[] lives in fixed VGPRs (no v_movrels / M0).
// ---------------------------------------------------------------------------
__global__ __launch_bounds__(128) void qkv_kernel(
    const float* __restrict__ x,
    const float* __restrict__ Wq, const float* __restrict__ bq,
    const float* __restrict__ Wk, const float* __restrict__ bk,
    const float* __restrict__ Wv, const float* __restrict__ bv,
    float* __restrict__ ws) {
  const int NT = (BB * SS) / 16;  // 512 row tiles per matrix
  int tid   = blockIdx.x * 4 + (threadIdx.x >> 5);
  int which = tid / NT;           // 0=Q 1=K 2=V
  int tile  = tid % NT;
  int r0    = tile * 16;

  const float* W    = (which == 0) ? Wq : (which == 1) ? Wk : Wv;
  const float* bias = (which == 0) ? bq : (which == 1) ? bk : bv;
  float* Out = ws + (size_t)which * (BB * SS * DD);

  int lane = threadIdx.x & 31;
  int half = lane >> 4;   // 0: K={0,1}, 1: K={2,3} in A fragment
  int l16  = lane & 15;

  // Preload all A fragments (the full 16x64 x-tile: 2 floats/lane/chunk)
  v2f afrag[16];
  const float* xrow = x + (size_t)(r0 + l16) * DD + 2 * half;
#pragma unroll
  for (int kc = 0; kc < 16; ++kc) {
    afrag[kc].x = xrow[4 * kc + 0];
    afrag[kc].y = xrow[4 * kc + 1];
  }

  v8f acc[4];
#pragma unroll
  for (int n = 0; n < 4; ++n)
#pragma unroll
    for (int v = 0; v < 8; ++v) acc[n][v] = 0.0f;

#pragma unroll
  for (int kc = 0; kc < 16; ++kc) {
    int krow = 4 * kc + 2 * half;
#pragma unroll
    for (int n = 0; n < 4; ++n) {
      v2f bfrag;
      bfrag.x = W[(size_t)krow * DD + n * 16 + l16];
      bfrag.y = W[(size_t)(krow + 1) * DD + n * 16 + l16];
      acc[n] = wmma_f32_16x16x4(afrag[kc], bfrag, acc[n]);
    }
  }

#pragma unroll
  for (int n = 0; n < 4; ++n) {
    int col = n * 16 + l16;
    float bb = bias[col];
#pragma unroll
    for (int v = 0; v < 8; ++v) {
      int row = r0 + v + 8 * half;   // C layout: M = v + 8*(lane>=16)
      Out[(size_t)row * DD + col] = fmaxf(acc[n][v] + bb, 0.0f);
    }
  }
}

// ---------------------------------------------------------------------------
// Kernel 2: flash attention per (b,h). One wave = 16 queries; key loop in
// blocks of 16. QK^T: 1 wmma (K=4). PV: 4 wmmas (N padded 4->16).
// Fused /l, residual; pre-BN result written to d_out.
// ---------------------------------------------------------------------------
__global__ __launch_bounds__(128) void attn_kernel(
    const float* __restrict__ x, const float* __restrict__ ws,
    float* __restrict__ out) {
  const float* Q = ws;
  const float* K = ws + (size_t)BB * SS * DD;
  const float* V = ws + (size_t)2 * BB * SS * DD;

  __shared__ float ldsP[4][16][16];  // per-wave P staging (C->A relayout)

  int wave = threadIdx.x >> 5;
  int lane = threadIdx.x & 31;
  int half = lane >> 4;
  int l16  = lane & 15;

  const int QT = SS / 64;  // 32 query-blocks of 64 per (b,h)
  int bh = blockIdx.x / QT;
  int qt = blockIdx.x % QT;
  int b = bh / HH, h = bh % HH;
  int q0 = qt * 64 + wave * 16;

  const float* Qb = Q + ((size_t)b * SS) * DD + h * HDIM;
  const float* Kb = K + ((size_t)b * SS) * DD + h * HDIM;
  const float* Vb = V + ((size_t)b * SS) * DD + h * HDIM;

  // Q fragment: A = [16 q x 4 hd]
  v2f aq;
  aq.x = Qb[(size_t)(q0 + l16) * DD + 2 * half];
  aq.y = Qb[(size_t)(q0 + l16) * DD + 2 * half + 1];

  float mrow[8], lrow[8];
  v8f accO, zc;
#pragma unroll
  for (int v = 0; v < 8; ++v) {
    mrow[v] = -INFINITY;
    lrow[v] = 0.0f;
    accO[v] = 0.0f;
    zc[v]   = 0.0f;
  }

  const float scale = 0.5f;  // 1/sqrt(HD)

  for (int k0 = 0; k0 < SS; k0 += 16) {
    if (k0 + 16 < SS) {
      __builtin_prefetch(Kb + (size_t)(k0 + 16 + l16) * DD, 0, 1);
      __builtin_prefetch(Vb + (size_t)(k0 + 16 + l16) * DD, 0, 1);
    }
    // K^T fragment: B = [4 hd x 16 keys], B[k][n] = K[k0+n][k]
    v2f bk;
    bk.x = Kb[(size_t)(k0 + l16) * DD + 2 * half];
    bk.y = Kb[(size_t)(k0 + l16) * DD + 2 * half + 1];
    v8f s = wmma_f32_16x16x4(aq, bk, zc);

    // online softmax per row (row = v + 8*half, 16 cols across lane group)
    float p[8];
#pragma unroll
    for (int v = 0; v < 8; ++v) {
      float sv = s[v] * scale;
      float m = sv;
      m = fmaxf(m, __shfl_xor(m, 1, 16));
      m = fmaxf(m, __shfl_xor(m, 2, 16));
      m = fmaxf(m, __shfl_xor(m, 4, 16));
      m = fmaxf(m, __shfl_xor(m, 8, 16));
      float mnew  = fmaxf(mrow[v], m);
      float alpha = __expf(mrow[v] - mnew);
      float pv    = __expf(sv - mnew);
      float rs = pv;
      rs += __shfl_xor(rs, 1, 16);
      rs += __shfl_xor(rs, 2, 16);
      rs += __shfl_xor(rs, 4, 16);
      rs += __shfl_xor(rs, 8, 16);
      lrow[v] = lrow[v] * alpha + rs;
      mrow[v] = mnew;
      accO[v] *= alpha;
      p[v] = pv;
    }

    // stage P (C layout) to LDS, read back in A layout. Wave-local: DS ops
    // from one wave are in-order; compiler inserts s_wait_dscnt for the loads.
#pragma unroll
    for (int v = 0; v < 8; ++v) ldsP[wave][v + 8 * half][l16] = p[v];

#pragma unroll
    for (int kc = 0; kc < 4; ++kc) {
      v2f ap;  // A = P[16 q x 4 keys]
      ap.x = ldsP[wave][l16][4 * kc + 2 * half];
      ap.y = ldsP[wave][l16][4 * kc + 2 * half + 1];
      v2f bv;  // B = V[4 keys x 16 cols] (cols >= HD are zero padding)
      int krow = k0 + 4 * kc + 2 * half;
      bv.x = (l16 < HDIM) ? Vb[(size_t)krow * DD + l16] : 0.0f;
      bv.y = (l16 < HDIM) ? Vb[(size_t)(krow + 1) * DD + l16] : 0.0f;
      accO = wmma_f32_16x16x4(ap, bv, accO);
    }
  }

  // epilogue: normalize, residual, write pre-BN output
  if (l16 < HDIM) {
    int ch = h * HDIM + l16;
#pragma unroll
    for (int v = 0; v < 8; ++v) {
      int row = q0 + v + 8 * half;
      size_t idx = ((size_t)b * SS + row) * DD + ch;
      out[idx] = accO[v] / lrow[v] + x[idx];
    }
  }
}

// ---------------------------------------------------------------------------
// Kernel 3: per-channel mean / rstd over B*S rows (one block per channel).
// ---------------------------------------------------------------------------
__global__ __launch_bounds__(256) void bn_stats_kernel(
    const float* __restrict__ out, float* __restrict__ stats) {
  __shared__ float ssum[256], ssq[256];
  int ch = blockIdx.x;
  int t = threadIdx.x;
  float s = 0.0f, q = 0.0f;
  for (int i = t; i < BB * SS; i += 256) {
    float v = out[(size_t)i * DD + ch];
    s += v;
    q += v * v;
  }
  ssum[t] = s;
  ssq[t] = q;
  __syncthreads();
  for (int off = 128; off > 0; off >>= 1) {
    if (t < off) {
      ssum[t] += ssum[t + off];
      ssq[t] += ssq[t + off];
    }
    __syncthreads();
  }
  if (t == 0) {
    const float invN = 1.0f / (float)(BB * SS);
    float mean = ssum[0] * invN;
    float var = ssq[0] * invN - mean * mean;  // biased variance
    stats[ch] = mean;
    stats[DD + ch] = rsqrtf(var + 1e-5f);
  }
}

// ---------------------------------------------------------------------------
// Kernel 4: in-place BN apply.
// ---------------------------------------------------------------------------
__global__ __launch_bounds__(256) void bn_apply_kernel(
    float* __restrict__ out, const float* __restrict__ stats,
    const float* __restrict__ gamma, const float* __restrict__ beta) {
  int idx = blockIdx.x * 256 + threadIdx.x;
  int ch = idx & (DD - 1);
  float v = out[idx];
  out[idx] = (v - stats[ch]) * stats[DD + ch] * gamma[ch] + beta[ch];
}

extern "C" void kernel_launch(void* const* d_in, const int* in_sizes, int n_in,
                              void* d_out, int out_size, void* d_ws,
                              size_t ws_size, hipStream_t stream) {
  const float* x     = (const float*)d_in[0];
  const float* Wq    = (const float*)d_in[1];
  const float* bq    = (const float*)d_in[2];
  const float* Wk    = (const float*)d_in[3];
  const float* bk    = (const float*)d_in[4];
  const float* Wv    = (const float*)d_in[5];
  const float* bv    = (const float*)d_in[6];
  const float* gamma = (const float*)d_in[7];
  const float* beta  = (const float*)d_in[8];
  float* out = (float*)d_out;
  float* ws = (float*)d_ws;                     // Q,K,V: 3 * 2MB
  float* stats = ws + (size_t)3 * BB * SS * DD; // 128 floats

  // 1536 tiles (3 matrices x 512 row-tiles), 4 waves/block
  qkv_kernel<<<(3 * (BB * SS / 16)) / 4, 128, 0, stream>>>(x, Wq, bq, Wk, bk,
                                                           Wv, bv, ws);
  // B*H*(S/64) = 2048 blocks, 4 waves each (16 queries per wave)
  attn_kernel<<<BB * HH * (SS / 64), 128, 0, stream>>>(x, ws, out);
  bn_stats_kernel<<<DD, 256, 0, stream>>>(out, stats);
  bn_apply_kernel<<<(BB * SS * DD) / 256, 256, 0, stream>>>(out, stats, gamma,
                                                            beta);
}